// FusionAwareInterp_37795712204988
// MI455X (gfx1250) — compile-verified
//
#include <hip/hip_runtime.h>
#include <hip/hip_bf16.h>
#include <math.h>

#define BS   4
#define C    64
#define H_   64
#define W_   64
#define HW   4096
#define NPT  4096
#define KNN  3
#define LRELU 0.1f

typedef __attribute__((ext_vector_type(2))) float v2f;
typedef __attribute__((ext_vector_type(8))) float v8f;
typedef int v4i __attribute__((vector_size(16)));
typedef __attribute__((address_space(1))) v4i gas_v4i;
typedef __attribute__((address_space(3))) v4i las_v4i;

// ---------------------------------------------------------------------------
// Async global->LDS helpers (gfx1250 GLOBAL_LOAD_ASYNC_TO_LDS_B128, ASYNCcnt).
// Guarded so the file still compiles on toolchains lacking the builtins.
// ---------------------------------------------------------------------------
#if __has_builtin(__builtin_amdgcn_global_load_async_to_lds_b128)
#define HAVE_ASYNC_LDS 1
#else
#define HAVE_ASYNC_LDS 0
#endif

__device__ __forceinline__ void lds_copy16(const float* __restrict__ g,
                                           float* __restrict__ l) {
#if HAVE_ASYNC_LDS
    __builtin_amdgcn_global_load_async_to_lds_b128(
        (gas_v4i*)g, (las_v4i*)l, 0, 0);
#else
    l[0] = g[0]; l[1] = g[1]; l[2] = g[2]; l[3] = g[3];
#endif
}

__device__ __forceinline__ void async_wait0() {
#if HAVE_ASYNC_LDS
#if __has_builtin(__builtin_amdgcn_s_wait_asynccnt)
    __builtin_amdgcn_s_wait_asynccnt(0);
#else
    asm volatile("s_wait_asynccnt 0x0" ::: "memory");
#endif
#endif
}

// ---------------------------------------------------------------------------
// Kernel 1: brute-force 3-NN.  One thread per query pixel; uv staged into LDS
// (32 KB of the 320 KB WGP pool) via async b128 copies, then a branchy top-3
// scan (wave-level update branch -> cheaper than a
// branchless cndmask insert).
// ---------------------------------------------------------------------------
__global__ void __launch_bounds__(256) knn_kernel(const float* __restrict__ uv,
                                                  int* __restrict__ knn_idx) {
    __shared__ float s_uv[2 * NPT];          // [x row | y row], mirrors global
    const int b   = blockIdx.x >> 4;         // 16 query-tiles of 256 per batch
    const int qt  = blockIdx.x & 15;
    const int tid = threadIdx.x;
    const float* uvb = uv + (size_t)b * 2 * NPT;

    // 32 KB contiguous: 2048 x 16B chunks, 8 per thread, async to LDS.
    #pragma unroll
    for (int i = 0; i < (2 * NPT / 4) / 256; ++i) {
        const int c4 = (i * 256 + tid) * 4;
        lds_copy16(uvb + c4, s_uv + c4);
    }
    async_wait0();
    __syncthreads();
    const float* s_ux = s_uv;
    const float* s_uy = s_uv + NPT;

    const int   q  = qt * 256 + tid;
    const float gx = (float)(q & (W_ - 1));
    const float gy = (float)(q >> 6);

    float d0 = 3.4e38f, d1 = 3.4e38f, d2 = 3.4e38f;
    int   i0 = 0, i1 = 0, i2 = 0;
    for (int n = 0; n < NPT; ++n) {
        float dx = s_ux[n] - gx;
        float dy = s_uy[n] - gy;
        float dd = fmaf(dx, dx, dy * dy);
        if (dd < d2) {
            if (dd < d1) {
                if (dd < d0) {
                    d2 = d1; i2 = i1;
                    d1 = d0; i1 = i0;
                    d0 = dd; i0 = n;
                } else {
                    d2 = d1; i2 = i1;
                    d1 = dd; i1 = n;
                }
            } else {
                d2 = dd; i2 = n;
            }
        }
    }
    int* o = knn_idx + ((size_t)b * HW + q) * KNN;
    o[0] = i0; o[1] = i1; o[2] = i2;
}

// ---------------------------------------------------------------------------
// Kernel 2: per-(b,q) scoring MLP (3->16->64, sigmoid) and weighted gather of
// feat_3d -> final (bs, C, HW).  w2 staged via async b128; broadcast LDS reads.
// ---------------------------------------------------------------------------
__global__ void __launch_bounds__(256) score_kernel(const float* __restrict__ uv,
                                                    const float* __restrict__ feat3d,
                                                    const float* __restrict__ w1,
                                                    const float* __restrict__ b1,
                                                    const float* __restrict__ w2,
                                                    const float* __restrict__ b2,
                                                    const int*   __restrict__ knn_idx,
                                                    float*       __restrict__ final_out) {
    __shared__ float s_w2[C * 16];           // 4 KB
    __shared__ float s_w1[16 * 3];
    __shared__ float s_b1[16];
    __shared__ float s_b2[C];
    const int tid = threadIdx.x;
    // 1024 floats = 256 lanes x 16B: single async b128 round.
    lds_copy16(w2 + tid * 4, s_w2 + tid * 4);
    if (tid < 16 * 3) s_w1[tid] = w1[tid];
    if (tid < 16)     s_b1[tid] = b1[tid];
    if (tid < C)      s_b2[tid] = b2[tid];
    async_wait0();
    __syncthreads();

    const int   b  = blockIdx.x >> 4;
    const int   q  = (blockIdx.x & 15) * 256 + tid;
    const float gx = (float)(q & (W_ - 1));
    const float gy = (float)(q >> 6);

    const float* uvb  = uv + (size_t)b * 2 * NPT;
    const int*   idxp = knn_idx + ((size_t)b * HW + q) * KNN;

    int   idx[KNN];
    float h[KNN][16];
    #pragma unroll
    for (int k = 0; k < KNN; ++k) {
        idx[k] = idxp[k];
        float ox = uvb[idx[k]]       - gx;
        float oy = uvb[NPT + idx[k]] - gy;
        float on = sqrtf(fmaf(ox, ox, oy * oy));
        #pragma unroll
        for (int j = 0; j < 16; ++j) {
            float v = s_b1[j];
            v = fmaf(s_w1[j * 3 + 0], ox, v);
            v = fmaf(s_w1[j * 3 + 1], oy, v);
            v = fmaf(s_w1[j * 3 + 2], on, v);
            h[k][j] = (v >= 0.f) ? v : LRELU * v;
        }
    }

    const float* f3 = feat3d + (size_t)b * C * NPT;
    float*       fo = final_out + (size_t)b * C * HW + q;
    for (int c = 0; c < C; ++c) {
        float acc = 0.f;
        #pragma unroll
        for (int k = 0; k < KNN; ++k) {
            float s = s_b2[c];
            #pragma unroll
            for (int j = 0; j < 16; ++j) s = fmaf(s_w2[c * 16 + j], h[k][j], s);
            float sc = 1.0f / (1.0f + __expf(-s));
            acc = fmaf(sc, f3[(size_t)c * NPT + idx[k]], acc);
        }
        fo[(size_t)c * HW] = acc;
    }
}

// ---------------------------------------------------------------------------
// Kernel 3: out = lrelu(w_out @ final + b_out): per batch a 64x64 * 64x4096
// fp32 GEMM on the matrix core.  Each wave owns one 16(M)x32(N) tile: the 16
// k-step A fragments of w_out are preloaded once into 32 VGPRs and reused
// across both N-subtiles -> 32 chained v_wmma_f32_16x16x4_f32 per wave.
// 4 waves/block cover M=64; 128 N-supertiles x 4 batches = 512 blocks.
// EXEC is all-ones throughout (WMMA requirement).
// ---------------------------------------------------------------------------
__global__ void __launch_bounds__(128) outconv_kernel(const float* __restrict__ w_out,
                                                      const float* __restrict__ b_out,
                                                      const float* __restrict__ final_in,
                                                      float*       __restrict__ out) {
    const int b    = blockIdx.x >> 7;   // 128 N-supertiles per batch
    const int nt   = blockIdx.x & 127;
    const int wave = threadIdx.x >> 5;
    const int lane = threadIdx.x & 31;
    const int o0   = wave * 16;         // M-tile origin (output channel)
    const int q0   = nt * 32;           // N-supertile origin (pixel)

    // ISA 16x4 f32 A layout: lane = M (0..15), lanes 16-31 hold K+2;
    // VGPR0 = K{0|2}, VGPR1 = K{1|3}.  B mirrors with N = lane&15.
    const int mrow  = lane & 15;
    const int khalf = (lane >> 4) * 2;

    // Preload all 16 A fragments (K=64) once; reused for both N-subtiles.
    v2f a[16];
    {
        const float* ap = w_out + (size_t)(o0 + mrow) * C + khalf;
        #pragma unroll
        for (int kk = 0; kk < 16; ++kk) {
            a[kk].x = ap[4 * kk + 0];
            a[kk].y = ap[4 * kk + 1];
        }
    }

    const float* fb = final_in + (size_t)b * C * HW;
    const int ncol  = lane & 15;
    const int mbase = (lane >> 4) * 8;
    float* ob = out + (size_t)b * C * HW;

    #pragma unroll
    for (int half = 0; half < 2; ++half) {
        const int qq = q0 + half * 16;
        v8f acc = {};
        #pragma unroll
        for (int kk = 0; kk < 16; ++kk) {
            v2f bm;
            const float* bp = fb + (size_t)(4 * kk + khalf) * HW + qq + mrow;
            bm.x = bp[0];
            bm.y = bp[HW];
            acc = __builtin_amdgcn_wmma_f32_16x16x4_f32(false, a[kk], false, bm,
                                                        (short)0, acc, false, false);
        }
        // D layout: VGPR i -> M = i + 8*(lane>=16), N = lane&15
        #pragma unroll
        for (int i = 0; i < 8; ++i) {
            const int m = o0 + i + mbase;
            float v = acc[i] + b_out[m];
            v = (v >= 0.f) ? v : LRELU * v;
            ob[(size_t)m * HW + qq + ncol] = v;
        }
    }
}

// ---------------------------------------------------------------------------
extern "C" void kernel_launch(void* const* d_in, const int* in_sizes, int n_in,
                              void* d_out, int out_size, void* d_ws, size_t ws_size,
                              hipStream_t stream) {
    const float* uv     = (const float*)d_in[0];
    // d_in[1] = feat_2d: the reference only uses its SHAPE, never its values.
    const float* feat3d = (const float*)d_in[2];
    const float* w1     = (const float*)d_in[3];
    const float* b1     = (const float*)d_in[4];
    const float* w2     = (const float*)d_in[5];
    const float* b2     = (const float*)d_in[6];
    const float* w_out  = (const float*)d_in[7];
    const float* b_out  = (const float*)d_in[8];
    float* out = (float*)d_out;

    char*  ws       = (char*)d_ws;
    int*   knn_idx  = (int*)ws;                                   // bs*HW*3 ints
    size_t idxBytes = ((size_t)BS * HW * KNN * sizeof(int) + 255) & ~(size_t)255;
    float* final_buf = (float*)(ws + idxBytes);                   // bs*C*HW floats

    knn_kernel<<<BS * (HW / 256), 256, 0, stream>>>(uv, knn_idx);
    score_kernel<<<BS * (HW / 256), 256, 0, stream>>>(uv, feat3d, w1, b1, w2, b2,
                                                      knn_idx, final_buf);
    outconv_kernel<<<BS * (HW / 32), 128, 0, stream>>>(w_out, b_out, final_buf, out);
}